// S4Cell_60378650247537
// MI455X (gfx1250) — compile-verified
//
#include <hip/hip_runtime.h>
#include <math.h>

#define N_STATE 64
#define H_CH    256
#define L_SEQ   4096
#define L2X     8192
#define LOG2_L  12
#define LOG2_L2 13

typedef __attribute__((ext_vector_type(2))) float v2f;
typedef __attribute__((ext_vector_type(8))) float v8f;

__device__ __forceinline__ unsigned bitrev(unsigned x, int bits) {
  return __brev(x) >> (32 - bits);
}

// v_rcp_f32 + one Newton-Raphson step (~1 ulp), avoids the IEEE div sequence.
__device__ __forceinline__ float frcp(float x) {
  float r = __builtin_amdgcn_rcpf(x);
  r = fmaf(fmaf(-x, r, 1.0f), r, r);
  return r;
}

// DIT radix-2 FFT in LDS: input bit-reversed order, output natural order.
__device__ void fft_dit(float* re, float* im, int n, float sign) {
  const int tid = threadIdx.x, nt = blockDim.x;
  for (int len = 2; len <= n; len <<= 1) {
    const int half = len >> 1;
    const float base = sign * 6.28318530717958647692f / (float)len;
    for (int i = tid; i < (n >> 1); i += nt) {
      const int j = i / half, k = i - j * half;
      const int p0 = j * len + k, p1 = p0 + half;
      float s, c;
      __sincosf(base * (float)k, &s, &c);
      const float xr = re[p1], xi = im[p1];
      const float tr = c * xr - s * xi, ti = c * xi + s * xr;
      const float ur = re[p0], ui = im[p0];
      re[p1] = ur - tr; im[p1] = ui - ti;
      re[p0] = ur + tr; im[p0] = ui + ti;
    }
    __syncthreads();
  }
}

// DIF radix-2 FFT in LDS: input natural order, output bit-reversed order.
__device__ void fft_dif(float* re, float* im, int n, float sign) {
  const int tid = threadIdx.x, nt = blockDim.x;
  for (int len = n; len >= 2; len >>= 1) {
    const int half = len >> 1;
    const float base = sign * 6.28318530717958647692f / (float)len;
    for (int i = tid; i < (n >> 1); i += nt) {
      const int j = i / half, k = i - j * half;
      const int p0 = j * len + k, p1 = p0 + half;
      float s, c;
      __sincosf(base * (float)k, &s, &c);
      const float ar = re[p0], ai = im[p0];
      const float br = re[p1], bi = im[p1];
      re[p0] = ar + br; im[p0] = ai + bi;
      const float dr = ar - br, di = ai - bi;
      re[p1] = c * dr - s * di;
      im[p1] = c * di + s * dr;
    }
    __syncthreads();
  }
}

extern "C" __global__ __launch_bounds__(256)
void s4_fused_kernel(const float* __restrict__ u,
                     const float* __restrict__ lam_re,
                     const float* __restrict__ lam_im,
                     const float* __restrict__ p_re,  const float* __restrict__ p_im,
                     const float* __restrict__ b_re,  const float* __restrict__ b_im,
                     const float* __restrict__ cmat,  const float* __restrict__ dvec,
                     const float* __restrict__ log_step,
                     float* __restrict__ out) {
  extern __shared__ float smem[];
  float* b1r = smem;              // 8192
  float* b1i = b1r + L2X;         // 8192
  float* b2r = b1i + L2X;         // 8192
  float* b2i = b2r + L2X;         // 8192
  float* sB  = b2i + L2X;         // 2048 : B matrix, 128 (K) x 16 (N)
  float* sC  = sB + 2048;         // 2048 : per-wave 16x16 C spill (8 waves x 256)
  float* sLr = sC + 2048;         // 64
  float* sLi = sLr + 64;          // 64

  const int h    = blockIdx.x;
  const int tid  = threadIdx.x;
  const int lane = tid & 31;
  const int wv   = tid >> 5;

  // ---- build shared weight matrix B (128x16) and pole tables ----
  for (int i = tid; i < 2048; i += 256) sB[i] = 0.f;
  if (tid < N_STATE) {
    const int n = tid;
    sLr[n] = fminf(lam_re[h * N_STATE + n], -1e-4f);
    sLi[n] = lam_im[h * N_STATE + n];
    const float ccr = cmat[(h * N_STATE + n) * 2 + 0];
    const float cci = cmat[(h * N_STATE + n) * 2 + 1];
    const float pr = p_re[h * N_STATE + n], pi = p_im[h * N_STATE + n];
    const float br = b_re[h * N_STATE + n], bi = b_im[h * N_STATE + n];
    float wr[4], wi[4];
    wr[0] = ccr * br + cci * bi;  wi[0] = ccr * bi - cci * br;  // conj(c)*b
    wr[1] = ccr * pr + cci * pi;  wi[1] = ccr * pi - cci * pr;  // conj(c)*p
    wr[2] = pr * br + pi * bi;    wi[2] = pr * bi - pi * br;    // conj(p)*b
    wr[3] = pr * pr + pi * pi;    wi[3] = 0.f;                  // conj(p)*p
    #pragma unroll
    for (int q = 0; q < 4; ++q) {
      sB[n * 16 + 2 * q]                 =  wr[q];
      sB[n * 16 + 2 * q + 1]             =  wi[q];
      sB[(N_STATE + n) * 16 + 2 * q]     = -wi[q];
      sB[(N_STATE + n) * 16 + 2 * q + 1] =  wr[q];
    }
  }
  const float stepinv2 = 2.f * __expf(-log_step[h]);  // 2/step
  __syncthreads();

  // ---- Cauchy kernel via V_WMMA_F32_16X16X4_F32 ----
  // A (16 l-rows x 128 K): [Re(1/(g-lam)) | Im(1/(g-lam))].
  // K-loop split: cols 0..63 compute recip once, stash Im parts in regs for cols 64..127.
  // C columns 0..7 = k00.re, k00.im, k01.re, k01.im, k10.re, k10.im, k11.re, k11.im
  const int row  = lane & 15;
  const int koff = (lane < 16) ? 0 : 2;
  for (int tile = wv; tile < L_SEQ / 16; tile += 8) {
    const int l = tile * 16 + row;
    float sn, cs;
    __sincosf(-6.28318530717958647692f * (float)l / (float)L_SEQ, &sn, &cs);
    const float dr = 1.f + cs, di = sn;                  // 1 + omega
    const float dinv = frcp(dr * dr + di * di);
    const float qr = (1.f - (cs * cs + sn * sn)) * dinv; // (1-w)/(1+w)
    const float qi = (-2.f * sn) * dinv;
    const float gr = stepinv2 * qr, gi = stepinv2 * qi;
    const float c2r = 2.f * dr * dinv, c2i = -2.f * di * dinv;  // 2/(1+w)

    v8f acc = {};
    float sim[32];                       // stashed Im(recip) parts (stays in VGPRs)
    #pragma unroll
    for (int kk = 0; kk < 16; ++kk) {
      const int k0 = kk * 4 + koff;      // 0..63: Re half of A
      v2f a;
      {
        const float ar_ = gr - sLr[k0], ai_ = gi - sLi[k0];
        const float inv = frcp(ar_ * ar_ + ai_ * ai_);
        a.x = ar_ * inv;
        sim[2 * kk] = -ai_ * inv;
      }
      {
        const float ar_ = gr - sLr[k0 + 1], ai_ = gi - sLi[k0 + 1];
        const float inv = frcp(ar_ * ar_ + ai_ * ai_);
        a.y = ar_ * inv;
        sim[2 * kk + 1] = -ai_ * inv;
      }
      v2f bf;
      bf.x = sB[k0 * 16 + row];
      bf.y = sB[(k0 + 1) * 16 + row];
      acc = __builtin_amdgcn_wmma_f32_16x16x4_f32(false, a, false, bf,
                                                  (short)0, acc, false, false);
    }
    #pragma unroll
    for (int kk = 0; kk < 16; ++kk) {
      const int k0 = 64 + kk * 4 + koff; // 64..127: Im half of A, replay stash
      v2f a;
      a.x = sim[2 * kk];
      a.y = sim[2 * kk + 1];
      v2f bf;
      bf.x = sB[k0 * 16 + row];
      bf.y = sB[(k0 + 1) * 16 + row];
      acc = __builtin_amdgcn_wmma_f32_16x16x4_f32(false, a, false, bf,
                                                  (short)0, acc, false, false);
    }

    // spill C tile so lanes can read full rows
    {
      float* cb = sC + wv * 256;
      const int Mb = (lane < 16) ? 0 : 8;
      #pragma unroll
      for (int r = 0; r < 8; ++r) cb[(Mb + r) * 16 + row] = acc[r];
    }
    __syncthreads();
    if (lane < 16) {
      const float* cb = sC + wv * 256 + lane * 16;
      const float k00r = cb[0], k00i = cb[1], k01r = cb[2], k01i = cb[3];
      const float k10r = cb[4], k10i = cb[5], k11r = cb[6], k11i = cb[7];
      const float er = 1.f + k11r, ei = k11i;
      const float einv = frcp(er * er + ei * ei);
      const float ir = er * einv, ii = -ei * einv;           // 1/(1+k11)
      const float m1r = k01r * ir - k01i * ii, m1i = k01r * ii + k01i * ir;
      const float tr = m1r * k10r - m1i * k10i, ti = m1r * k10i + m1i * k10r;
      const float vr = k00r - tr, vi = k00i - ti;
      const float rr = c2r * vr - c2i * vi, ri = c2r * vi + c2i * vr;
      const unsigned pos = bitrev((unsigned)l, LOG2_L);      // pre-permute for DIT
      b1r[pos] = rr; b1i[pos] = ri;
    }
    __syncthreads();
  }

  // ---- kker = ifft_4096(roots).real ----
  fft_dit(b1r, b1i, L_SEQ, +1.f);

  // ---- Kd = fft_8192(pad(kker)) : scatter bit-reversed into buf2, DIT ----
  const float sc1 = 1.f / (float)L_SEQ;
  for (int j = tid; j < L2X; j += 256) {
    const unsigned pj = bitrev((unsigned)j, LOG2_L2);
    b2r[pj] = (j < L_SEQ) ? (b1r[j] * sc1) : 0.f;
    b2i[pj] = 0.f;
  }
  __syncthreads();
  fft_dit(b2r, b2i, L2X, -1.f);   // Kd, natural order, stays in buf2

  // ---- Fu = fft_8192(pad(u[:,h])) ----
  for (int j = tid; j < L2X; j += 256) {
    const unsigned pj = bitrev((unsigned)j, LOG2_L2);
    b1r[pj] = (j < L_SEQ) ? u[j * H_CH + h] : 0.f;
    b1i[pj] = 0.f;
  }
  __syncthreads();
  fft_dit(b1r, b1i, L2X, -1.f);   // Fu, natural order

  // ---- pointwise product (in place, natural order: no staging needed) ----
  for (int j = tid; j < L2X; j += 256) {
    const float fr = b1r[j], fi = b1i[j];
    const float kr = b2r[j], ki = b2i[j];
    b1r[j] = fr * kr - fi * ki;
    b1i[j] = fr * ki + fi * kr;
  }
  __syncthreads();

  // ---- inverse FFT via DIF (natural in -> bit-reversed out), gather at store ----
  fft_dif(b1r, b1i, L2X, +1.f);
  const float sc2 = 1.f / (float)L2X;
  const float dh = dvec[h];
  for (int l = tid; l < L_SEQ; l += 256) {
    const unsigned pl = bitrev((unsigned)l, LOG2_L2);
    out[l * H_CH + h] = b1r[pl] * sc2 + u[l * H_CH + h] * dh;
  }
}

extern "C" void kernel_launch(void* const* d_in, const int* in_sizes, int n_in,
                              void* d_out, int out_size, void* d_ws, size_t ws_size,
                              hipStream_t stream) {
  (void)in_sizes; (void)n_in; (void)out_size; (void)d_ws; (void)ws_size;
  const float* u   = (const float*)d_in[0];
  const float* lr  = (const float*)d_in[1];
  const float* li  = (const float*)d_in[2];
  const float* pr  = (const float*)d_in[3];
  const float* pi  = (const float*)d_in[4];
  const float* br  = (const float*)d_in[5];
  const float* bi  = (const float*)d_in[6];
  const float* cc  = (const float*)d_in[7];
  const float* dd  = (const float*)d_in[8];
  const float* ls  = (const float*)d_in[9];
  float* out = (float*)d_out;
  const size_t shmem = (size_t)(4 * L2X + 2048 + 2048 + 128) * sizeof(float); // ~145 KB
  hipLaunchKernelGGL(s4_fused_kernel, dim3(H_CH), dim3(256), shmem, stream,
                     u, lr, li, pr, pi, br, bi, cc, dd, ls, out);
}